// ModelNew_25056839204936
// MI455X (gfx1250) — compile-verified
//
#include <hip/hip_runtime.h>

// Problem constants (reference: IN_FEATURES = OUT_FEATURES = 4096, batch 4096)
#define IFEAT 4096
#define OFEAT 4096
#define BATCH 4096
#define NCHUNK 32                    // row-chunks for deterministic colsum tree
#define ROWS_PER_CHUNK (OFEAT / NCHUNK)

typedef float v2f __attribute__((ext_vector_type(2)));
typedef float v8f __attribute__((ext_vector_type(8)));

// ---------------------------------------------------------------------------
// Phase 1a: partial column sums of W.  grid = (4, NCHUNK), block = 256.
// Each thread owns 4 consecutive columns (one float4) -> fully coalesced
// 512B-per-wave GLOBAL_LOAD_B128 stream over 64 MB of W, read exactly once.
// ---------------------------------------------------------------------------
__global__ void __launch_bounds__(256)
colsum_partial_kernel(const float* __restrict__ W, float* __restrict__ part) {
    const int col4 = blockIdx.x * 256 + threadIdx.x;   // float4 column index [0,1024)
    const int r0   = blockIdx.y * ROWS_PER_CHUNK;
    const float4* W4 = (const float4*)W;
    float4 acc = make_float4(0.f, 0.f, 0.f, 0.f);
    #pragma unroll 4
    for (int r = 0; r < ROWS_PER_CHUNK; ++r) {
        float4 w = W4[(size_t)(r0 + r) * (IFEAT / 4) + col4];
        acc.x += w.x; acc.y += w.y; acc.z += w.z; acc.w += w.w;
    }
    ((float4*)part)[(size_t)blockIdx.y * (IFEAT / 4) + col4] = acc;
}

// Phase 1b: deterministic reduction of the NCHUNK partials. grid = 4, block = 256.
__global__ void __launch_bounds__(256)
colsum_reduce_kernel(const float* __restrict__ part, float* __restrict__ wsum) {
    const int col4 = blockIdx.x * 256 + threadIdx.x;
    const float4* p4 = (const float4*)part;
    float4 acc = make_float4(0.f, 0.f, 0.f, 0.f);
    #pragma unroll
    for (int c = 0; c < NCHUNK; ++c) {
        float4 v = p4[(size_t)c * (IFEAT / 4) + col4];
        acc.x += v.x; acc.y += v.y; acc.z += v.z; acc.w += v.w;
    }
    ((float4*)wsum)[col4] = acc;
}

// Fallback (small workspace): direct column sum, one column per thread.
__global__ void __launch_bounds__(256)
colsum_direct_kernel(const float* __restrict__ W, float* __restrict__ wsum) {
    const int col = blockIdx.x * 256 + threadIdx.x;    // [0, 4096)
    float acc = 0.f;
    for (int r = 0; r < OFEAT; ++r)
        acc += W[(size_t)r * IFEAT + col];
    wsum[col] = acc;
}

// Bias sum: single block, deterministic LDS tree.
__global__ void __launch_bounds__(256)
bias_sum_kernel(const float* __restrict__ b, float* __restrict__ bsum) {
    __shared__ float s[256];
    float a = 0.f;
    for (int i = threadIdx.x; i < OFEAT; i += 256) a += b[i];
    s[threadIdx.x] = a;
    __syncthreads();
    for (int w = 128; w > 0; w >>= 1) {
        if (threadIdx.x < w) s[threadIdx.x] += s[threadIdx.x + w];
        __syncthreads();
    }
    if (threadIdx.x == 0) *bsum = s[0];
}

// ---------------------------------------------------------------------------
// Phase 2: out[r] = dot(x[r], wsum) + bsum via V_WMMA_F32_16X16X4_F32.
//   A (16x4 f32): lane = M (lane&15), lane-half selects K base {0,2}; 2 VGPRs.
//   B (4x16 f32): wsum chunk broadcast across all N columns -> per-lane value
//                 depends only on K: VGPR0 = v[k+0]/v[k+2], VGPR1 = v[k+1]/v[k+3].
//   D: we consume column N=0 only: lane 0 holds M=0..7, lane 16 holds M=8..15.
// Each 16-row tile is K-split over 4 waves (1024 K each, 256 WMMAs), partials
// combined deterministically through LDS. 128 blocks x 8 waves = 1024 waves.
// ---------------------------------------------------------------------------
__global__ void __launch_bounds__(256)
matvec_wmma_kernel(const float* __restrict__ x,
                   const float* __restrict__ wsum,
                   const float* __restrict__ bsum_p,
                   float* __restrict__ out) {
    __shared__ float red[2][4][16];

    const int lane   = threadIdx.x & 31;
    const int wave   = threadIdx.x >> 5;       // 0..7
    const int tloc   = wave >> 2;              // tile within block: 0..1
    const int ksplit = wave & 3;               // 0..3
    const int row0   = (blockIdx.x * 2 + tloc) * 16;

    const int m     = lane & 15;               // A-matrix row
    const int khalf = (lane >> 4) * 2;         // K base within group of 4: 0 or 2

    const float*  xrow = x + (size_t)(row0 + m) * IFEAT + khalf;
    const float4* v4   = (const float4*)wsum;

    v8f c = {0.f, 0.f, 0.f, 0.f, 0.f, 0.f, 0.f, 0.f};

    const int k4beg = ksplit * (IFEAT / 4 / 4);        // 256 float4-groups per wave
    const int k4end = k4beg + (IFEAT / 4 / 4);
    #pragma unroll 4
    for (int k4 = k4beg; k4 < k4end; ++k4) {
        v2f a = *(const v2f*)(xrow + (size_t)k4 * 4);  // A tile fragment (b64 load)
        float4 vv = v4[k4];                            // broadcast wsum chunk
        v2f bm;
        bm.x = (lane < 16) ? vv.x : vv.z;
        bm.y = (lane < 16) ? vv.y : vv.w;
        // D = A(16x4) * B(4x16) + C, exact f32
        c = __builtin_amdgcn_wmma_f32_16x16x4_f32(
                /*neg_a=*/false, a, /*neg_b=*/false, bm,
                /*c_mod=*/(short)0, c, /*reuse_a=*/false, /*reuse_b=*/false);
    }

    // Column N=0 of D: lane 0 -> rows m=0..7 (VGPR 0..7), lane 16 -> rows 8..15.
    if (lane == 0) {
        #pragma unroll
        for (int r = 0; r < 8; ++r) red[tloc][ksplit][r] = c[r];
    } else if (lane == 16) {
        #pragma unroll
        for (int r = 0; r < 8; ++r) red[tloc][ksplit][8 + r] = c[r];
    }
    __syncthreads();

    // Deterministic combine of the 4 K-split partials; 32 rows per block.
    if (threadIdx.x < 32) {
        const int t = threadIdx.x >> 4;
        const int r = threadIdx.x & 15;
        float s = red[t][0][r] + red[t][1][r] + red[t][2][r] + red[t][3][r];
        out[blockIdx.x * 32 + threadIdx.x] = s + *bsum_p;
    }
}

// ---------------------------------------------------------------------------
extern "C" void kernel_launch(void* const* d_in, const int* in_sizes, int n_in,
                              void* d_out, int out_size, void* d_ws, size_t ws_size,
                              hipStream_t stream) {
    (void)in_sizes; (void)n_in; (void)out_size;
    const float* x = (const float*)d_in[0];
    const float* W = (const float*)d_in[1];
    const float* b = (const float*)d_in[2];
    float* out = (float*)d_out;

    float* wsum = (float*)d_ws;                  // IFEAT floats
    float* bsum = wsum + IFEAT;                  // 1 float (padded to 4)
    float* part = wsum + IFEAT + 4;              // NCHUNK*IFEAT floats, 16B aligned

    const size_t need_big = (size_t)(IFEAT + 4 + (size_t)NCHUNK * IFEAT) * sizeof(float);

    if (ws_size >= need_big) {
        // Deterministic two-phase column-sum tree (preferred: 128 + 4 blocks).
        colsum_partial_kernel<<<dim3(4, NCHUNK), 256, 0, stream>>>(W, part);
        colsum_reduce_kernel<<<dim3(4), 256, 0, stream>>>(part, wsum);
    } else {
        // Small-workspace fallback: direct column sum.
        colsum_direct_kernel<<<dim3(16), 256, 0, stream>>>(W, wsum);
    }
    bias_sum_kernel<<<1, 256, 0, stream>>>(b, bsum);

    // 256 row-tiles of 16, each K-split across 4 waves -> 128 blocks x 8 waves.
    matvec_wmma_kernel<<<dim3(128), 256, 0, stream>>>(x, wsum, bsum, out);
}